// GAT_33638183862785
// MI455X (gfx1250) — compile-verified
//
#include <hip/hip_runtime.h>

typedef float v2f __attribute__((ext_vector_type(2)));
typedef float v8f __attribute__((ext_vector_type(8)));

#define BB   4
#define NN   1024
#define HH   128
#define FF   128
#define EE   16
#define GG   128
#define OO   128
#define DIN  256
#define LRELU 0.01f

__device__ __forceinline__ v8f wmma_f32(v2f a, v2f b, v8f c) {
  // V_WMMA_F32_16X16X4_F32: D = A(16x4) * B(4x16) + C(16x16), fp32
  return __builtin_amdgcn_wmma_f32_16x16x4_f32(false, a, false, b, (short)0, c,
                                               false, false);
}

// values is stored K-pair interleaved: element (row j, col o) lives at
// float2 index (j>>1)*OO + o, component (j&1). Row j = b*N + n.
// This makes each WMMA B-fragment a single aligned b64 load.

// ---------------------------------------------------------------------------
// Kernel 1: per 16-row tile: values = feat@W_m + b_m  (-> ws, packed)
//                            skip   = feat@W_skip + b_skip (-> d_out)
//                            att1/att2 row projections (-> ws)
// ---------------------------------------------------------------------------
__global__ __launch_bounds__(256) void gat_proj(
    const float* __restrict__ hidden, const float* __restrict__ nfeat,
    const float* __restrict__ Wm, const float* __restrict__ bm,
    const float* __restrict__ Wsk, const float* __restrict__ bsk,
    const float* __restrict__ wa1, const float* __restrict__ ba1,
    const float* __restrict__ wa2, const float* __restrict__ ba2,
    float2* __restrict__ values_pk, float* __restrict__ att1,
    float* __restrict__ att2, float* __restrict__ out_skip) {
  __shared__ float sf[16 * (DIN + 8)];  // 16 rows x 256, stride 264 (bank pad)
  const int t = threadIdx.x;
  const int r0 = blockIdx.x * 16;  // global row = b*N + n

  // stage concat(n_features, hidden) tile into LDS
  for (int idx = t; idx < 16 * DIN; idx += 256) {
    const int i = idx >> 8;
    const int k = idx & 255;
    const int r = r0 + i;
    const float v = (k < FF) ? nfeat[r * FF + k] : hidden[r * HH + (k - FF)];
    sf[i * (DIN + 8) + k] = v;
  }
  __syncthreads();

  // att1/att2: 16 threads per row, shfl-16 reduction
  {
    const int i = t >> 4, l = t & 15;
    float s1 = 0.f, s2 = 0.f;
    const float* fr = &sf[i * (DIN + 8)];
    for (int k = l; k < DIN; k += 16) {
      const float f = fr[k];
      s1 += f * wa1[k];
      s2 += f * wa2[k];
    }
    for (int m = 8; m >= 1; m >>= 1) {
      s1 += __shfl_xor(s1, m, 16);
      s2 += __shfl_xor(s2, m, 16);
    }
    if (l == 0) {
      att1[r0 + i] = s1 + ba1[0];
      att2[r0 + i] = s2 + ba2[0];
    }
  }

  // WMMA: wave w owns output columns [16w, 16w+16) for both W_m and W_skip
  const int lane = t & 31;
  const int wv = t >> 5;
  const int M = lane & 15;
  const int off = (lane >> 4) << 1;      // K offset 0 or 2
  const int col = wv * 16 + (lane & 15); // output column
  v8f accm = {};
  v8f accs = {};
  const float* fr = &sf[M * (DIN + 8)];
  for (int kb = 0; kb < DIN; kb += 4) {
    v2f a;
    a.x = fr[kb + off];
    a.y = fr[kb + off + 1];
    v2f b1;
    b1.x = Wm[(kb + off) * OO + col];
    b1.y = Wm[(kb + off + 1) * OO + col];
    accm = wmma_f32(a, b1, accm);
    v2f b2;
    b2.x = Wsk[(kb + off) * OO + col];
    b2.y = Wsk[(kb + off + 1) * OO + col];
    accs = wmma_f32(a, b2, accs);
  }
  const float biasm = bm[col], biass = bsk[col];
  const int rbase = r0 + ((lane >> 4) << 3);  // D row = v + 8*(lane>=16); even
#pragma unroll
  for (int v = 0; v < 8; v += 2) {
    const int r = rbase + v;
    float2 pk;
    pk.x = accm[v] + biasm;
    pk.y = accm[v + 1] + biasm;
    values_pk[(r >> 1) * OO + col] = pk;  // one b64 store per row pair
    out_skip[r * OO + col] = accs[v] + biass;
    out_skip[(r + 1) * OO + col] = accs[v + 1] + biass;
  }
}

// ---------------------------------------------------------------------------
// Kernel 2: att_g[b] = g_features[b,:] . w_ag + b_ag
// ---------------------------------------------------------------------------
__global__ __launch_bounds__(128) void gat_attg(
    const float* __restrict__ g, const float* __restrict__ wag,
    const float* __restrict__ bag, float* __restrict__ attg) {
  __shared__ float red[4];
  const int b = blockIdx.x, t = threadIdx.x;
  float v = g[b * GG + t] * wag[t];
  for (int m = 16; m >= 1; m >>= 1) v += __shfl_xor(v, m, 32);
  if ((t & 31) == 0) red[t >> 5] = v;
  __syncthreads();
  if (t == 0) attg[b] = red[0] + red[1] + red[2] + red[3] + bag[0];
}

// ---------------------------------------------------------------------------
// Kernel 3: fused edge-scores + masked softmax + coefs@values, one block per
// (b, 16-row tile). Scores live in LDS (16 x 1032 floats, bank-padded).
// ---------------------------------------------------------------------------
__global__ __launch_bounds__(256) void gat_attn(
    const float* __restrict__ efeat, const float* __restrict__ adj,
    const float* __restrict__ wae, const float* __restrict__ bae,
    const float* __restrict__ att1, const float* __restrict__ att2,
    const float* __restrict__ attg, const float2* __restrict__ values_pk,
    float* __restrict__ out) {
  __shared__ float sc[16 * (NN + 8)];  // 64.5 KB of the 320 KB LDS
  const int t = threadIdx.x;
  const int b = blockIdx.x >> 6;    // 64 row-tiles per batch
  const int i0 = (blockIdx.x & 63) * 16;

  float ae[EE];
#pragma unroll
  for (int e = 0; e < EE; e++) ae[e] = wae[e];
  const float base_add = bae[0] + attg[b];

  // ---- score + softmax pass: 16 threads per row, 64 edges per thread ----
  {
    const int i = t >> 4, tj = t & 15;
    const int ri = b * NN + i0 + i;
    const float a1 = att1[ri] + base_add;
    const float* erow = efeat + (size_t)ri * NN * EE;
    const float* arow = adj + (size_t)ri * NN;
    const float* a2row = att2 + b * NN;
    float* srow = &sc[i * (NN + 8)];
    float mx = -3.0e38f;
    for (int j = tj; j < NN; j += 16) {
      const float4* ep = (const float4*)(erow + j * EE);
      const float4 e0 = ep[0], e1 = ep[1], e2 = ep[2], e3 = ep[3];
      const float d = e0.x * ae[0] + e0.y * ae[1] + e0.z * ae[2] + e0.w * ae[3] +
                      e1.x * ae[4] + e1.y * ae[5] + e1.z * ae[6] + e1.w * ae[7] +
                      e2.x * ae[8] + e2.y * ae[9] + e2.z * ae[10] + e2.w * ae[11] +
                      e3.x * ae[12] + e3.y * ae[13] + e3.z * ae[14] + e3.w * ae[15];
      const float lg = a1 + a2row[j] + d;
      const float lr = lg > 0.f ? lg : LRELU * lg;          // leaky_relu
      const float s = lr + (arow[j] - 1.0f) * 1.0e9f;       // adjacency mask
      srow[j] = s;
      mx = fmaxf(mx, s);
    }
    for (int m = 8; m >= 1; m >>= 1) mx = fmaxf(mx, __shfl_xor(mx, m, 16));
    float sum = 0.f;
    for (int j = tj; j < NN; j += 16) {
      const float e = __expf(srow[j] - mx);
      srow[j] = e;
      sum += e;
    }
    for (int m = 8; m >= 1; m >>= 1) sum += __shfl_xor(sum, m, 16);
    const float inv = 1.0f / sum;
    for (int j = tj; j < NN; j += 16) srow[j] *= inv;
  }
  __syncthreads();

  // ---- coefs[16x1024] @ values[1024x128] via f32 WMMA ----
  const int lane = t & 31;
  const int wv = t >> 5;
  const int M = lane & 15;
  const int off = (lane >> 4) << 1;   // even K offset -> aligned pair
  const int col = wv * 16 + (lane & 15);
  const float* crow = &sc[M * (NN + 8)];
  const float2* vb = values_pk + ((size_t)b * NN >> 1) * OO;
  v8f acc = {};
  for (int kb = 0; kb < NN; kb += 4) {
    v2f a;
    a.x = crow[kb + off];
    a.y = crow[kb + off + 1];
    const float2 bv = vb[((kb + off) >> 1) * OO + col];  // single b64 load
    v2f bf;
    bf.x = bv.x;
    bf.y = bv.y;
    acc = wmma_f32(a, bf, acc);
  }
  const int rb = i0 + ((lane >> 4) << 3);
#pragma unroll
  for (int v = 0; v < 8; v++) {
    const size_t idx = ((size_t)b * NN + rb + v) * OO + col;
    out[idx] += acc[v];  // add onto skip term already in d_out
  }
}

// ---------------------------------------------------------------------------
extern "C" void kernel_launch(void* const* d_in, const int* in_sizes, int n_in,
                              void* d_out, int out_size, void* d_ws,
                              size_t ws_size, hipStream_t stream) {
  const float* hidden = (const float*)d_in[0];
  const float* nfeat = (const float*)d_in[1];
  const float* efeat = (const float*)d_in[2];
  const float* gfeat = (const float*)d_in[3];
  const float* adj = (const float*)d_in[4];
  const float* Wm = (const float*)d_in[5];
  const float* bm = (const float*)d_in[6];
  const float* Wsk = (const float*)d_in[7];
  const float* bsk = (const float*)d_in[8];
  const float* wa1 = (const float*)d_in[9];
  const float* ba1 = (const float*)d_in[10];
  const float* wa2 = (const float*)d_in[11];
  const float* ba2 = (const float*)d_in[12];
  const float* wae = (const float*)d_in[13];
  const float* bae = (const float*)d_in[14];
  const float* wag = (const float*)d_in[15];
  const float* bag = (const float*)d_in[16];
  float* out = (float*)d_out;

  float* ws = (float*)d_ws;
  float2* values_pk = (float2*)ws;              // B*N*O floats (packed pairs)
  float* att1 = ws + BB * NN * OO;              // B*N
  float* att2 = att1 + BB * NN;                 // B*N
  float* attg = att2 + BB * NN;                 // B

  gat_proj<<<BB * NN / 16, 256, 0, stream>>>(hidden, nfeat, Wm, bm, Wsk, bsk,
                                             wa1, ba1, wa2, ba2, values_pk,
                                             att1, att2, out);
  gat_attg<<<BB, 128, 0, stream>>>(gfeat, wag, bag, attg);
  gat_attn<<<BB * (NN / 16), 256, 0, stream>>>(efeat, adj, wae, bae, att1,
                                               att2, attg, values_pk, out);
}